// DigitCaps_17403207483947
// MI455X (gfx1250) — compile-verified
//
#include <hip/hip_runtime.h>

// DigitCaps dynamic routing, fused for MI455X (gfx1250).
// Strategy: never materialize u_hat (335 MB); W(42MB)+u(17MB) stay resident in
// the 192 MB L2, u_hat tiles are recomputed per routing pass with
// V_WMMA_F32_16X16X4_F32 (full fp32 precision, matrix pipe).
// Workspace layout (floats): [logits B*N*C][s B*C*O][v B*C*O][c B*N*C] = 21.3 MB.

constexpr int B     = 128;
constexpr int NCAPS = 2048;
constexpr int DIM   = 16;
constexpr int C     = 10;
constexpr int O     = 32;
constexpr float EPS = 1e-8f;

typedef __attribute__((ext_vector_type(2))) float v2f;
typedef __attribute__((ext_vector_type(8))) float v8f;

__device__ __forceinline__ v8f wmma_f32_16x16x4(v2f a, v2f b, v8f c) {
  // D(16x16,f32) = A(16x4,f32) * B(4x16,f32) + C
  return __builtin_amdgcn_wmma_f32_16x16x4_f32(
      /*neg_a=*/false, a, /*neg_b=*/false, b,
      /*c_mod=*/(short)0, c, /*reuse_a=*/false, /*reuse_b=*/false);
}

// ---------------------------------------------------------------- utilities
__global__ void zero_f32(float* __restrict__ p, int n) {
  int i = blockIdx.x * blockDim.x + threadIdx.x;
  if (i < n) p[i] = 0.f;
}

// c[b,n,:] = softmax(logits[b,n,:]) over the 10 classes. One thread per (b,n).
__global__ void softmax_c(const float* __restrict__ logits,
                          float* __restrict__ cbuf) {
  int i = blockIdx.x * blockDim.x + threadIdx.x;  // flat (b,n), exact grid
  const float* lp = logits + i * C;
  float l[C];
  float m = -3.4e38f;
#pragma unroll
  for (int k = 0; k < C; ++k) { l[k] = lp[k]; m = fmaxf(m, l[k]); }
  float s = 0.f;
#pragma unroll
  for (int k = 0; k < C; ++k) { l[k] = __expf(l[k] - m); s += l[k]; }
  float inv = 1.f / s;
  float* cp = cbuf + i * C;
#pragma unroll
  for (int k = 0; k < C; ++k) cp[k] = l[k] * inv;
}

// ------------------------------------------------ s accumulation (WMMA GEMM)
// Per class: s[b,o] += sum_{k=(n,d)} (c[b,n]*u[b,n,d]) * W[n,cls,o,d]
// Wave = (btile 0..7, cls 0..9, nseg 0..31 of 64 n's). K chunked by 4 (one n
// contributes 4 K-chunks). One A fragment feeds both o-tiles (N=32).
__global__ void __launch_bounds__(128)
s_accum(const float* __restrict__ u, const float* __restrict__ W,
        const float* __restrict__ cbuf, float* __restrict__ sbuf) {
  const int lane  = threadIdx.x & 31;
  const int wid   = blockIdx.x * 4 + (threadIdx.x >> 5);
  const int btile = wid & 7;
  const int cls   = (wid >> 3) % C;
  const int nseg  = wid / (8 * C);

  const int mb = lane & 15;            // A: M = b_local; B: N = o_local
  const int kk = (lane >> 4) * 2;      // lane-half K offset (ISA 16x4 layout)
  const int b  = btile * 16 + mb;

  v8f acc0 = {}; v8f acc1 = {};
  const int n0 = nseg * 64;
  for (int n = n0; n < n0 + 64; ++n) {
    const float cval = cbuf[(b * NCAPS + n) * C + cls];
    const float* up  = u + (b * NCAPS + n) * DIM;
    const float* wp  = W + (size_t)(n * C + cls) * O * DIM;
#pragma unroll
    for (int ch = 0; ch < 4; ++ch) {
      const int d0 = ch * 4 + kk;                       // K = n*16 + d
      v2f a = *(const v2f*)(up + d0);
      a *= cval;                                        // A = c * u
      v2f b0 = *(const v2f*)(wp + mb * DIM + d0);        // o-tile 0
      v2f b1 = *(const v2f*)(wp + (mb + 16) * DIM + d0); // o-tile 1
      acc0 = wmma_f32_16x16x4(a, b0, acc0);
      acc1 = wmma_f32_16x16x4(a, b1, acc1);
    }
  }
  // D layout: vgpr r -> M=r (lanes 0-15) / M=r+8 (lanes 16-31), N=lane&15.
  const int hi8 = (lane >> 4) * 8;
#pragma unroll
  for (int r = 0; r < 8; ++r) {
    int bb = btile * 16 + r + hi8;
    atomicAdd(&sbuf[(bb * C + cls) * O + mb],      acc0[r]);
    atomicAdd(&sbuf[(bb * C + cls) * O + 16 + mb], acc1[r]);
  }
}

// ------------------------------------------------------------------- squash
// v = |s|^2/(1+|s|^2) * s/(|s|+eps); also re-zeros s for the next pass.
__global__ void squash_k(float* __restrict__ sbuf, float* __restrict__ vout) {
  int bc  = blockIdx.x;            // b*C + cls
  int o   = threadIdx.x;           // 0..31
  int idx = bc * O + o;
  float x  = sbuf[idx];
  float ss = x * x;
#pragma unroll
  for (int m = 16; m; m >>= 1) ss += __shfl_xor(ss, m, 32);
  float n2    = ss + EPS;
  float norm  = sqrtf(n2);
  float scale = n2 / (1.f + n2);
  vout[idx] = scale * x / (norm + EPS);
  sbuf[idx] = 0.f;
}

// --------------------------------------- agreement / logits update (WMMA)
// t[b,d] = sum_o v[b,cls,o] * W[n,cls,o,d]   (GEMM M=16b, K=32o, N=16d)
// logits[b,n,cls] += sum_d t[b,d] * u[b,n,d] (half-wave shuffle reduce)
__global__ void __launch_bounds__(128)
agreement_k(const float* __restrict__ u, const float* __restrict__ W,
            const float* __restrict__ vbuf, float* __restrict__ logits) {
  const int lane  = threadIdx.x & 31;
  const int wid   = blockIdx.x * 4 + (threadIdx.x >> 5);
  const int btile = wid & 7;
  const int cls   = (wid >> 3) % C;
  const int n     = wid / (8 * C);

  const int mb = lane & 15;           // A: M = b_local; B: N = d
  const int kk = (lane >> 4) * 2;
  const int b  = btile * 16 + mb;

  const float* vp = vbuf + (b * C + cls) * O;
  const float* wp = W + (size_t)(n * C + cls) * O * DIM;

  v8f acc = {};
#pragma unroll
  for (int ch = 0; ch < 8; ++ch) {
    const int o0 = ch * 4 + kk;        // K = o
    v2f a = *(const v2f*)(vp + o0);
    v2f bf;
    bf[0] = wp[o0 * DIM + mb];
    bf[1] = wp[(o0 + 1) * DIM + mb];
    acc = wmma_f32_16x16x4(a, bf, acc);
  }
  const int hi8 = (lane >> 4) * 8;
#pragma unroll
  for (int r = 0; r < 8; ++r) {
    int bb  = btile * 16 + r + hi8;
    float t = acc[r] * u[(bb * NCAPS + n) * DIM + mb];
#pragma unroll
    for (int m = 8; m; m >>= 1) t += __shfl_xor(t, m, 32);  // reduce d in half
    if (mb == 0) logits[(bb * NCAPS + n) * C + cls] += t;   // unique writer
  }
}

// ------------------------------------------------------------------- launch
extern "C" void kernel_launch(void* const* d_in, const int* in_sizes, int n_in,
                              void* d_out, int out_size, void* d_ws,
                              size_t ws_size, hipStream_t stream) {
  const float* u = (const float*)d_in[0];  // [B, N, D]
  const float* W = (const float*)d_in[1];  // [N, C, O, D]
  float* out = (float*)d_out;              // [B, C, O]
  float* wsf = (float*)d_ws;

  float* logits = wsf;                         // B*N*C
  float* sbuf   = logits + B * NCAPS * C;      // B*C*O (adjacent: one zero pass)
  float* vbuf   = sbuf + B * C * O;            // B*C*O
  float* cbuf   = vbuf + B * C * O;            // B*N*C

  const int zn = B * NCAPS * C + B * C * O;    // logits + s contiguous
  zero_f32<<<(zn + 255) / 256, 256, 0, stream>>>(logits, zn);

  for (int it = 0; it < 3; ++it) {
    softmax_c<<<B * NCAPS / 256, 256, 0, stream>>>(logits, cbuf);
    // 8 btiles * 10 classes * 32 nsegs = 2560 waves = 640 blocks of 4 waves
    s_accum<<<640, 128, 0, stream>>>(u, W, cbuf, sbuf);
    float* vdst = (it == 2) ? out : vbuf;
    squash_k<<<B * C, 32, 0, stream>>>(sbuf, vdst);
    if (it < 2) {
      // 2048 n * 10 classes * 8 btiles = 163840 waves = 40960 blocks
      agreement_k<<<40960, 128, 0, stream>>>(u, W, vbuf, logits);
    }
  }
}